// DistSelfAttention_39195871543883
// MI455X (gfx1250) — compile-verified
//
#include <hip/hip_runtime.h>
#include <hip/hip_bf16.h>
#include <math.h>

// ---------------------------------------------------------------------------
// Distributional self-attention for MI455X (gfx1250, wave32, WMMA).
// All matrix work on V_WMMA_F32_16X16X4_F32 (full fp32 — kernel is HBM-bound
// by the 537 MB eps stream, so there is no reason to drop precision).
// ---------------------------------------------------------------------------

#define B_ 4
#define S_ 512
#define D_ 1024
#define H_ 16
#define HD_ 64
#define KTOP 8
#define MC_ 8
#define EPS6f 1e-6f

typedef __attribute__((ext_vector_type(2))) float v2f;
typedef __attribute__((ext_vector_type(8))) float v8f;

__device__ __forceinline__ v8f wmma4(v2f a, v2f b, v8f c) {
  // D = A(16x4,f32) * B(4x16,f32) + C(16x16,f32)
  return __builtin_amdgcn_wmma_f32_16x16x4_f32(false, a, false, b, (short)0, c,
                                               false, false);
}

// ---------------------------------------------------------------------------
// Projection: Omu = Xloc @ W^T + bias ; var = Xscale^2 @ (W^2)^T
// mode 0 (q/k/v): Osc = (sqrt(var)+1e-6)^2   (effective variance after split)
// mode 1 (out):   Osc = sqrt(var)            (output scale)
// One wave computes one 16x16 tile of the 2048x1024 output.
// ---------------------------------------------------------------------------
__global__ __launch_bounds__(256) void proj_kernel(
    const float* __restrict__ Xloc, const float* __restrict__ Xscale,
    const float* __restrict__ W, const float* __restrict__ bias,
    float* __restrict__ Omu, float* __restrict__ Osc, int mode) {
  const int gtid = blockIdx.x * blockDim.x + threadIdx.x;
  const int wave = gtid >> 5;
  const int lane = gtid & 31;
  const int hl = lane >> 4;  // half-wave select
  const int r = lane & 15;
  const int NT = D_ / 16;
  const int tm = wave / NT, tn = wave % NT;
  const int m0 = tm * 16, n0 = tn * 16;

  const float* arow = Xloc + (size_t)(m0 + r) * D_;
  const float* srow = Xscale + (size_t)(m0 + r) * D_;
  const float* wrow = W + (size_t)(n0 + r) * D_;

  v8f accm = {0.f, 0.f, 0.f, 0.f, 0.f, 0.f, 0.f, 0.f};
  v8f accv = {0.f, 0.f, 0.f, 0.f, 0.f, 0.f, 0.f, 0.f};
  for (int k0 = 0; k0 < D_; k0 += 4) {
    const int ka = k0 + 2 * hl;
    const float x0 = arow[ka], x1 = arow[ka + 1];
    const float s0 = srow[ka], s1 = srow[ka + 1];
    const float w0 = wrow[ka], w1 = wrow[ka + 1];
    v2f a, a2, bm, bv;
    a[0] = x0;       a[1] = x1;
    a2[0] = s0 * s0; a2[1] = s1 * s1;
    bm[0] = w0;      bm[1] = w1;
    bv[0] = w0 * w0; bv[1] = w1 * w1;
    accm = wmma4(a, bm, accm);
    accv = wmma4(a2, bv, accv);
  }
  const float bcol = bias[n0 + r];
#pragma unroll
  for (int j = 0; j < 8; ++j) {
    const int mrow = m0 + j + 8 * hl;
    const size_t o = (size_t)mrow * D_ + (n0 + r);
    Omu[o] = accm[j] + bcol;
    const float sc = sqrtf(accv[j]);
    Osc[o] = (mode == 0) ? (sc + EPS6f) * (sc + EPS6f) : sc;
  }
}

// ---------------------------------------------------------------------------
// Scores: per (b,h) tile of (q,k):
//   S_mu  = qm.km ;  S_var = qm^2.kv + qv.(km^2 + kv)
//   l_mu  = S_mu/(8*tau) ; l_var = ((S_var+e)/64 + e)/tau^2 + e
// Three WMMA chains per k-step over HD=64.
// ---------------------------------------------------------------------------
__global__ __launch_bounds__(256) void score_kernel(
    const float* __restrict__ qmu, const float* __restrict__ qvar,
    const float* __restrict__ kmu, const float* __restrict__ kvar,
    const float* __restrict__ taup, float* __restrict__ Lmu,
    float* __restrict__ Lvar) {
  const int gtid = blockIdx.x * blockDim.x + threadIdx.x;
  int w = gtid >> 5;
  const int lane = gtid & 31;
  const int hl = lane >> 4;
  const int r = lane & 15;
  const int tk = w & 31; w >>= 5;
  const int tq = w & 31; w >>= 5;
  const int h = w & 15;  w >>= 4;
  const int b = w;
  const int q0 = tq * 16, k0 = tk * 16;

  const size_t qbase = (size_t)(b * S_ + q0 + r) * D_ + h * HD_;
  const size_t kbase = (size_t)(b * S_ + k0 + r) * D_ + h * HD_;

  v8f am = {0.f, 0.f, 0.f, 0.f, 0.f, 0.f, 0.f, 0.f};
  v8f av = {0.f, 0.f, 0.f, 0.f, 0.f, 0.f, 0.f, 0.f};
  for (int d0 = 0; d0 < HD_; d0 += 4) {
    const int da = d0 + 2 * hl;
    const float qm0 = qmu[qbase + da], qm1 = qmu[qbase + da + 1];
    const float qv0 = qvar[qbase + da], qv1 = qvar[qbase + da + 1];
    const float km0 = kmu[kbase + da], km1 = kmu[kbase + da + 1];
    const float kv0 = kvar[kbase + da], kv1 = kvar[kbase + da + 1];
    v2f A, Bf;
    A[0] = qm0; A[1] = qm1; Bf[0] = km0; Bf[1] = km1;
    am = wmma4(A, Bf, am);                          // qm . km
    A[0] = qm0 * qm0; A[1] = qm1 * qm1; Bf[0] = kv0; Bf[1] = kv1;
    av = wmma4(A, Bf, av);                          // qm^2 . kv
    A[0] = qv0; A[1] = qv1;
    Bf[0] = km0 * km0 + kv0; Bf[1] = km1 * km1 + kv1;
    av = wmma4(A, Bf, av);                          // qv . (km^2 + kv)
  }
  const float tau = *taup;
  const float cmu = 1.0f / (8.0f * tau);
  const float ct2 = 1.0f / (tau * tau);
#pragma unroll
  for (int j = 0; j < 8; ++j) {
    const int qq = q0 + j + 8 * hl;
    const size_t o = ((size_t)(b * H_ + h) * S_ + qq) * S_ + (k0 + r);
    Lmu[o] = am[j] * cmu;
    Lvar[o] = ((av[j] + EPS6f) * (1.0f / (float)HD_) + EPS6f) * ct2 + EPS6f;
  }
}

// ---------------------------------------------------------------------------
// Top-k(8) per score row; one wave per row, 8 shfl-xor argmax passes.
// Ties resolved toward the lower column index (matches jax.lax.top_k).
// ---------------------------------------------------------------------------
__global__ __launch_bounds__(256) void topk_kernel(const float* __restrict__ Lmu,
                                                   int* __restrict__ topk_out) {
  const int gtid = blockIdx.x * blockDim.x + threadIdx.x;
  const int row = gtid >> 5;
  const int lane = gtid & 31;
  const float* p = Lmu + (size_t)row * S_;
  float vals[16];
#pragma unroll
  for (int t = 0; t < 16; ++t) vals[t] = p[t * 32 + lane];
#pragma unroll
  for (int sel = 0; sel < KTOP; ++sel) {
    float v = -3.402823466e38f;
    int vi = 0x7fffffff;
#pragma unroll
    for (int t = 0; t < 16; ++t) {
      const float x = vals[t];
      const int c = t * 32 + lane;
      if (x > v || (x == v && c < vi)) { v = x; vi = c; }
    }
    for (int off = 16; off; off >>= 1) {
      const float ov = __shfl_xor(v, off);
      const int oi = __shfl_xor(vi, off);
      if (ov > v || (ov == v && oi < vi)) { v = ov; vi = oi; }
    }
    const int slot = vi >> 5;
    if (lane == (vi & 31)) {
#pragma unroll
      for (int t = 0; t < 16; ++t)
        if (t == slot) vals[t] = -3.402823466e38f;
    }
    if (lane == 0) topk_out[row * KTOP + sel] = vi;
  }
}

// ---------------------------------------------------------------------------
// MC softmax + logit-normal rest + combine. One wave per row; 16 cols/lane
// (stride-32 => fully coalesced eps streaming, the dominant HBM traffic).
// Writes attn_mu in-place over Lmu.
// ---------------------------------------------------------------------------
__global__ __launch_bounds__(256) void attn_kernel(
    float* __restrict__ Lmu, const float* __restrict__ Lvar,
    const int* __restrict__ topk_in, const float* __restrict__ eps) {
  const int gtid = blockIdx.x * blockDim.x + threadIdx.x;
  const int row = gtid >> 5;
  const int lane = gtid & 31;
  float* rowp = Lmu + (size_t)row * S_;
  const float* varp = Lvar + (size_t)row * S_;

  int idxs[KTOP];
#pragma unroll
  for (int i = 0; i < KTOP; ++i) idxs[i] = topk_in[row * KTOP + i];

  float lmu[16], sd[16], rest[16], pmean[16];
  unsigned mbits = 0;
#pragma unroll
  for (int t = 0; t < 16; ++t) {
    const int c = t * 32 + lane;
    const float m = rowp[c];
    const float v = varp[c];
    lmu[t] = m;
    sd[t] = sqrtf(fmaxf(v, 1e-9f));
    float adj = m * rsqrtf(1.0f + 0.3926990816987241f * v);  // pi/8
    bool sel = false;
#pragma unroll
    for (int i = 0; i < KTOP; ++i) sel = sel || (c == idxs[i]);
    if (sel) { mbits |= (1u << t); adj = -1e9f; }
    rest[t] = adj;
    pmean[t] = 0.0f;
  }
  // logit-normal "rest" softmax
  float mx = rest[0];
#pragma unroll
  for (int t = 1; t < 16; ++t) mx = fmaxf(mx, rest[t]);
  for (int off = 16; off; off >>= 1) mx = fmaxf(mx, __shfl_xor(mx, off));
  float sum = 0.0f;
#pragma unroll
  for (int t = 0; t < 16; ++t) { rest[t] = __expf(rest[t] - mx); sum += rest[t]; }
  for (int off = 16; off; off >>= 1) sum += __shfl_xor(sum, off);
  const float rinv = 1.0f / sum;
#pragma unroll
  for (int t = 0; t < 16; ++t) rest[t] *= rinv;

  // 8 MC softmax samples (streams eps once)
  const size_t estride = (size_t)B_ * H_ * S_ * S_;
  const size_t ebase = (size_t)row * S_;
  for (int mc = 0; mc < MC_; ++mc) {
    const float* ep = eps + (size_t)mc * estride + ebase;
    float e[16];
    float m2 = -3.402823466e38f;
#pragma unroll
    for (int t = 0; t < 16; ++t) {
      e[t] = lmu[t] + sd[t] * ep[t * 32 + lane];
      m2 = fmaxf(m2, e[t]);
    }
    for (int off = 16; off; off >>= 1) m2 = fmaxf(m2, __shfl_xor(m2, off));
    float s2 = 0.0f;
#pragma unroll
    for (int t = 0; t < 16; ++t) { e[t] = __expf(e[t] - m2); s2 += e[t]; }
    for (int off = 16; off; off >>= 1) s2 += __shfl_xor(s2, off);
    const float wgt = (1.0f / (float)MC_) / s2;
#pragma unroll
    for (int t = 0; t < 16; ++t) pmean[t] += e[t] * wgt;
  }

  // top_out = where(mask, p_mean, rest); normalize; attn = top_out + rest
  float tsum = 0.0f;
  float tv[16];
#pragma unroll
  for (int t = 0; t < 16; ++t) {
    tv[t] = ((mbits >> t) & 1u) ? pmean[t] : rest[t];
    tsum += tv[t];
  }
  for (int off = 16; off; off >>= 1) tsum += __shfl_xor(tsum, off);
  const float dn = 1.0f / fmaxf(tsum, 1e-12f);
#pragma unroll
  for (int t = 0; t < 16; ++t) rowp[t * 32 + lane] = tv[t] * dn + rest[t];
}

// ---------------------------------------------------------------------------
// Y_mu = attn @ vm ; Y_var = attn^2 @ vv ; y_scale = sqrt(Y_var+e)+e
// Stores directly in combined [B,S,D] layout. Two WMMA chains per k-step.
// ---------------------------------------------------------------------------
__global__ __launch_bounds__(256) void av_kernel(
    const float* __restrict__ attn, const float* __restrict__ vmu,
    const float* __restrict__ vvar, float* __restrict__ yloc,
    float* __restrict__ ysc) {
  const int gtid = blockIdx.x * blockDim.x + threadIdx.x;
  int w = gtid >> 5;
  const int lane = gtid & 31;
  const int hl = lane >> 4;
  const int r = lane & 15;
  const int td = w & 3;  w >>= 2;  // HD/16 = 4 tiles
  const int tq = w & 31; w >>= 5;
  const int h = w & 15;  w >>= 4;
  const int b = w;
  const int q0 = tq * 16, d0 = td * 16;

  const size_t abase = ((size_t)(b * H_ + h) * S_ + q0 + r) * S_;
  const size_t vcol = (size_t)h * HD_ + d0 + r;

  v8f am = {0.f, 0.f, 0.f, 0.f, 0.f, 0.f, 0.f, 0.f};
  v8f av = {0.f, 0.f, 0.f, 0.f, 0.f, 0.f, 0.f, 0.f};
  for (int k0 = 0; k0 < S_; k0 += 4) {
    const int ka = k0 + 2 * hl;
    const float a0 = attn[abase + ka], a1 = attn[abase + ka + 1];
    const float b0 = vmu[(size_t)(b * S_ + ka) * D_ + vcol];
    const float b1 = vmu[(size_t)(b * S_ + ka + 1) * D_ + vcol];
    const float c0 = vvar[(size_t)(b * S_ + ka) * D_ + vcol];
    const float c1 = vvar[(size_t)(b * S_ + ka + 1) * D_ + vcol];
    v2f A, Bf;
    A[0] = a0; A[1] = a1; Bf[0] = b0; Bf[1] = b1;
    am = wmma4(A, Bf, am);
    A[0] = a0 * a0; A[1] = a1 * a1; Bf[0] = c0; Bf[1] = c1;
    av = wmma4(A, Bf, av);
  }
#pragma unroll
  for (int j = 0; j < 8; ++j) {
    const int qq = q0 + j + 8 * hl;
    const size_t o = (size_t)(b * S_ + qq) * D_ + vcol;
    yloc[o] = am[j];
    ysc[o] = sqrtf(av[j] + EPS6f) + EPS6f;
  }
}

// ---------------------------------------------------------------------------
extern "C" void kernel_launch(void* const* d_in, const int* in_sizes, int n_in,
                              void* d_out, int out_size, void* d_ws,
                              size_t ws_size, hipStream_t stream) {
  (void)in_sizes; (void)n_in; (void)out_size; (void)ws_size;
  const float* Qloc   = (const float*)d_in[0];
  const float* Qscale = (const float*)d_in[1];
  const float* Kloc   = (const float*)d_in[2];
  const float* Kscale = (const float*)d_in[3];
  const float* Vloc   = (const float*)d_in[4];
  const float* Vscale = (const float*)d_in[5];
  const float* Wq = (const float*)d_in[6];
  const float* bq = (const float*)d_in[7];
  const float* Wk = (const float*)d_in[8];
  const float* bk = (const float*)d_in[9];
  const float* Wv = (const float*)d_in[10];
  const float* bv = (const float*)d_in[11];
  const float* Wo = (const float*)d_in[12];
  const float* bo = (const float*)d_in[13];
  const float* tau = (const float*)d_in[14];
  const float* eps = (const float*)d_in[15];

  float* ws = (float*)d_ws;
  const size_t nTok = (size_t)B_ * S_ * D_;        // 2,097,152
  const size_t nAttn = (size_t)B_ * H_ * S_ * S_;  // 16,777,216
  float* q_mu  = ws;
  float* q_var = q_mu + nTok;
  float* k_mu  = q_var + nTok;
  float* k_var = k_mu + nTok;
  float* v_mu  = k_var + nTok;
  float* v_var = v_mu + nTok;
  float* l_mu  = v_var + nTok;
  float* l_var = l_mu + nAttn;
  int* topk_buf = (int*)(l_var + nAttn);
  float* y_loc = q_mu;   // reuse: q moments dead after score_kernel
  float* y_sc  = q_var;
  float* out_loc = (float*)d_out;
  float* out_sc  = out_loc + nTok;

  const dim3 blk(256);  // 8 wave32 per block

  // Q/K/V projections: (2048/16)*(1024/16) = 8192 waves -> 1024 blocks each
  proj_kernel<<<dim3(1024), blk, 0, stream>>>(Qloc, Qscale, Wq, bq, q_mu, q_var, 0);
  proj_kernel<<<dim3(1024), blk, 0, stream>>>(Kloc, Kscale, Wk, bk, k_mu, k_var, 0);
  proj_kernel<<<dim3(1024), blk, 0, stream>>>(Vloc, Vscale, Wv, bv, v_mu, v_var, 0);
  // Scores: B*H*32*32 = 65536 waves -> 8192 blocks
  score_kernel<<<dim3(8192), blk, 0, stream>>>(q_mu, q_var, k_mu, k_var, tau,
                                               l_mu, l_var);
  // Top-k: B*H*S = 32768 rows (one wave each) -> 4096 blocks
  topk_kernel<<<dim3(4096), blk, 0, stream>>>(l_mu, topk_buf);
  // MC + rest softmax combine (attn written over l_mu) -> 4096 blocks
  attn_kernel<<<dim3(4096), blk, 0, stream>>>(l_mu, l_var, topk_buf, eps);
  // attn @ V moments: B*H*32*4 = 8192 waves -> 1024 blocks
  av_kernel<<<dim3(1024), blk, 0, stream>>>(l_mu, v_mu, v_var, y_loc, y_sc);
  // Output projection (mode 1: scale = sqrt(var))
  proj_kernel<<<dim3(1024), blk, 0, stream>>>(y_loc, y_sc, Wo, bo, out_loc,
                                              out_sc, 1);
}